// VocalPitchLSTM_35811437314405
// MI455X (gfx1250) — compile-verified
//
#include <hip/hip_runtime.h>
#include <hip/hip_bf16.h>

#define SS 2
#define WIN 5
#define FS 360
#define UNITS 64
#define CH 64
#define TT 4096
#define FSP 368              // FS padded to 16
#define NGATES 256           // 4*UNITS
#define NMT (FSP / 16)       // 23 m-tiles
#define NTASK (NMT * 4)      // 92 (m-tile, unit-group) tasks

typedef float v2f __attribute__((ext_vector_type(2)));
typedef float v8f __attribute__((ext_vector_type(8)));

#if __has_builtin(__builtin_amdgcn_tanhf)
#define VP_TANH(v) __builtin_amdgcn_tanhf(v)
#elif __has_builtin(__builtin_amdgcn_tanh_f32)
#define VP_TANH(v) __builtin_amdgcn_tanh_f32(v)
#else
#define VP_TANH(v) tanhf(v)
#endif

__device__ __forceinline__ float vp_sigmoid(float v) {
    return 1.f / (1.f + __expf(-v));
}

// Weight LDS layout: K-pair interleaved so one ds_load_b64 yields a whole
// WMMA B fragment. Element (k, col) -> [(k>>1)*512 + col*2 + (k&1)].
__device__ __forceinline__ int vp_wpair(int k, int col) {
    return ((k >> 1) << 9) + (col << 1) + (k & 1);
}

// ---------------------------------------------------------------------------
// Kernel 1: peak-bin correlation (fully parallel, one thread per (t,f)).
// ---------------------------------------------------------------------------
__global__ __launch_bounds__(256) void vp_bins_kernel(const float* __restrict__ x,
                                                      int* __restrict__ bins) {
    int idx = blockIdx.x * blockDim.x + threadIdx.x;   // idx = t*FS + f
    if (idx >= TT * FS) return;
    int t = idx / FS;
    int f = idx - t * FS;

    float corr[WIN] = {0.f, 0.f, 0.f, 0.f, 0.f};
    if (t > 0) {
        const float4* cur = (const float4*)(x + (size_t)idx * CH);
        const float4* prow[WIN];
#pragma unroll
        for (int w = 0; w < WIN; ++w) {
            int j = f + w - SS;                        // reflect pad (no edge repeat)
            j = (j < 0) ? -j : ((j > FS - 1) ? (2 * (FS - 1) - j) : j);
            prow[w] = (const float4*)(x + ((size_t)(t - 1) * FS + j) * CH);
        }
        for (int c4 = 0; c4 < CH / 4; ++c4) {
            float4 xc = cur[c4];
#pragma unroll
            for (int w = 0; w < WIN; ++w) {
                float4 p = prow[w][c4];
                corr[w] += p.x * xc.x + p.y * xc.y + p.z * xc.z + p.w * xc.w;
            }
        }
    }
    if (f == 0)      { corr[0] -= 1.f; corr[1] -= 1.f; }
    if (f == 1)      { corr[0] -= 1.f; }
    if (f == FS - 2) { corr[WIN - 1] -= 1.f; }
    if (f == FS - 1) { corr[WIN - 2] -= 1.f; corr[WIN - 1] -= 1.f; }

    int best = 0;
    float bv = corr[0];
#pragma unroll
    for (int w = 1; w < WIN; ++w)                      // first-max wins (== jnp.argmax)
        if (corr[w] > bv) { bv = corr[w]; best = w; }
    int b = best + f - SS;
    bins[idx] = (b < 0) ? 0 : ((b > FS - 1) ? FS - 1 : b);
}

// ---------------------------------------------------------------------------
// Kernel 2 (optional, needs ~1.5GB ws): xz[t] = x[t] @ Wk + bias, fully
// parallel WMMA GEMM over all T. Removes the input GEMM from the serial chain.
// ---------------------------------------------------------------------------
__global__ __launch_bounds__(256) void vp_xz_kernel(const float* __restrict__ x,
                                                    const float* __restrict__ Wk,
                                                    const float* __restrict__ bias,
                                                    float* __restrict__ xz) {
    extern __shared__ float smem[];
    float* sW = smem;                 // CH*NGATES, pair layout
    float* sB = sW + CH * NGATES;     // NGATES

    const int tid = threadIdx.x;
    const int t = blockIdx.x;
    for (int i = tid; i < CH * NGATES; i += 256) {
        int k = i >> 8, col = i & 255;
        sW[vp_wpair(k, col)] = Wk[i];
    }
    if (tid < NGATES) sB[tid] = bias[tid];
    __syncthreads();

    const int wave = tid >> 5, lane = tid & 31;
    const int lhalf = lane >> 4, l16 = lane & 15;

    for (int task = wave; task < NMT * 16; task += 8) {      // 23 m-tiles x 16 n-tiles
        const int mt = task >> 4, nt = task & 15;
        const int m0 = mt * 16, n0 = nt * 16;
        const int arow = min(m0 + l16, FS - 1);
        const float* xrow = x + ((size_t)t * FS + arow) * CH;
        const int col = n0 + l16;

        v8f acc = {};
#pragma unroll 4
        for (int kk = 0; kk < CH / 4; ++kk) {
            const int k = kk * 4 + 2 * lhalf;
            const v2f ax = *(const v2f*)(xrow + k);
            const v2f bx = *(const v2f*)(sW + vp_wpair(k, col));
            acc = __builtin_amdgcn_wmma_f32_16x16x4_f32(
                false, ax, false, bx, (short)0, acc, false, false);
        }
        const float bv = sB[col];
#pragma unroll
        for (int v = 0; v < 8; ++v) {
            const int row = m0 + v + 8 * lhalf;
            if (row < FS)
                xz[((size_t)t * FS + row) * NGATES + col] = acc[v] + bv;
        }
    }
}

// ---------------------------------------------------------------------------
// Kernel 3: persistent single-workgroup LSTM; state h,c + recurrent weights
// resident in LDS. USE_XZ: read precomputed xz (serial chain = h-GEMM only);
// else fuse x @ Wk into the step. 2 barriers per step; bins double-buffered.
// ---------------------------------------------------------------------------
template <bool USE_XZ>
__global__ __launch_bounds__(1024, 1)
void vp_lstm_kernel(const float* __restrict__ xin,     // USE_XZ ? xz[T,FS,4U] : x[T,FS,CH]
                    const float* __restrict__ Wk,      // [CH, 4U] (fused only)
                    const float* __restrict__ Wr,      // [U, 4U]
                    const float* __restrict__ bias,    // [4U]    (fused only)
                    const int*   __restrict__ bins,    // [T, FS]
                    float* __restrict__ out) {         // [T, FS, U]
    extern __shared__ float smem[];
    float* sH  = smem;                                   // FSP*64 (rows >=360 written, never gathered)
    float* sC  = sH + FSP * UNITS;                       // FSP*64
    float* sWr = sC + FSP * UNITS;                       // 64*256 pair layout
    float* sWk = sWr + UNITS * NGATES;                   // 64*256 pair layout (fused only)
    float* sB  = USE_XZ ? sWk : (sWk + CH * NGATES);     // 256
    int*  sBin0 = (int*)(sB + NGATES);                   // FSP (double-buffered)
    int*  sBin1 = sBin0 + FSP;                           // FSP

    const int tid  = threadIdx.x;
    const int wave = tid >> 5, lane = tid & 31;
    const int lhalf = lane >> 4, l16 = lane & 15;

    for (int i = tid; i < FSP * UNITS; i += 1024) { sH[i] = 0.f; sC[i] = 0.f; }
    for (int i = tid; i < UNITS * NGATES; i += 1024) {
        int k = i >> 8, col = i & 255;
        int d = vp_wpair(k, col);
        sWr[d] = Wr[i];
        if (!USE_XZ) sWk[d] = Wk[i];
    }
    if (tid < NGATES) sB[tid] = bias[tid];
    for (int i = tid; i < FSP; i += 1024)                // bins row for t = 0
        sBin0[i] = bins[min(i, FS - 1)];
    __syncthreads();

    for (int t = 0; t < TT; ++t) {
        int* sBin  = (t & 1) ? sBin1 : sBin0;
        int* sBinN = (t & 1) ? sBin0 : sBin1;

        // ---- phase 1: GEMMs + pointwise (reads sH/sC/sBin only); results
        // staged in registers (task loop fully unrolled -> no scratch). ----
        float hn[3][8], cn[3][8];
#pragma unroll
        for (int s = 0; s < 3; ++s) {
            const int task = wave + 32 * s;
            if (task < NTASK) {                          // wave-uniform branch
                const int mt = task >> 2, gu = task & 3;
                const int m0 = mt * 16, u0 = gu * 16;
                const int arow = min(m0 + l16, FS - 1);
                const int rbin = sBin[arow];
                const float* hrow = sH + rbin * UNITS;
                const float* xrow = USE_XZ ? nullptr
                                           : xin + ((size_t)t * FS + arow) * CH;

                v8f acc[4] = {};
#pragma unroll 2
                for (int kk = 0; kk < UNITS / 4; ++kk) {
                    const int k = kk * 4 + 2 * lhalf;
                    const v2f ah = *(const v2f*)(hrow + k);
                    v2f ax;
                    if (!USE_XZ) ax = *(const v2f*)(xrow + k);
#pragma unroll
                    for (int g = 0; g < 4; ++g) {
                        const int col = g * UNITS + u0 + l16;
                        const v2f br = *(const v2f*)(sWr + vp_wpair(k, col));
                        acc[g] = __builtin_amdgcn_wmma_f32_16x16x4_f32(
                            false, ah, false, br, (short)0, acc[g], false, false);
                        if (!USE_XZ) {
                            const v2f bx = *(const v2f*)(sWk + vp_wpair(k, col));
                            acc[g] = __builtin_amdgcn_wmma_f32_16x16x4_f32(
                                false, ax, false, bx, (short)0, acc[g], false, false);
                        }
                    }
                }
                // warm L2/L0 for the next timestep's operands (global_prefetch_b8)
                {
                    const int tn = min(t + 1, TT - 1);
                    const size_t nb = USE_XZ
                        ? (((size_t)tn * FS + arow) * NGATES + u0 + l16)
                        : (((size_t)tn * FS + arow) * CH);
                    __builtin_prefetch(xin + nb, 0, 0);
                }

                const int u = u0 + l16;
                float bi, bf, bg, bo;
                if (!USE_XZ) {
                    bi = sB[0 * UNITS + u]; bf = sB[1 * UNITS + u];
                    bg = sB[2 * UNITS + u]; bo = sB[3 * UNITS + u];
                }
#pragma unroll
                for (int v = 0; v < 8; ++v) {
                    const int row  = m0 + v + 8 * lhalf;
                    const int rowc = min(row, FS - 1);
                    const float cold = sC[sBin[rowc] * UNITS + u];
                    float zi, zf, zg, zo;
                    if (USE_XZ) {
                        const float* zr = xin + ((size_t)t * FS + rowc) * NGATES;
                        zi = acc[0][v] + zr[0 * UNITS + u];
                        zf = acc[1][v] + zr[1 * UNITS + u];
                        zg = acc[2][v] + zr[2 * UNITS + u];
                        zo = acc[3][v] + zr[3 * UNITS + u];
                    } else {
                        zi = acc[0][v] + bi; zf = acc[1][v] + bf;
                        zg = acc[2][v] + bg; zo = acc[3][v] + bo;
                    }
                    const float cnew = vp_sigmoid(zf) * cold
                                     + vp_sigmoid(zi) * VP_TANH(zg);
                    cn[s][v] = cnew;
                    hn[s][v] = vp_sigmoid(zo) * VP_TANH(cnew);
                }
            }
        }
        __syncthreads();   // all gathered reads of sH/sC/sBin complete

        // ---- phase 2: commit state, stream output (NT store), and fetch the
        // next step's bins row into the alternate buffer. ----
        if (t + 1 < TT) {
            for (int i = tid; i < FSP; i += 1024)
                sBinN[i] = bins[(size_t)(t + 1) * FS + min(i, FS - 1)];
        }
#pragma unroll
        for (int s = 0; s < 3; ++s) {
            const int task = wave + 32 * s;
            if (task < NTASK) {
                const int mt = task >> 2, gu = task & 3;
                const int m0 = mt * 16;
                const int u  = gu * 16 + l16;
#pragma unroll
                for (int v = 0; v < 8; ++v) {
                    const int row = m0 + v + 8 * lhalf;
                    // LDS state padded to FSP rows: store unguarded (rows
                    // 360..367 are never gathered back, bins <= 359).
                    sH[row * UNITS + u] = hn[s][v];
                    sC[row * UNITS + u] = cn[s][v];
                    if (row < FS)      // streamed once, never re-read -> NT
                        __builtin_nontemporal_store(
                            hn[s][v], out + ((size_t)t * FS + row) * UNITS + u);
                }
            }
        }
        __syncthreads();   // state + next bins visible before next step
    }
}

// ---------------------------------------------------------------------------
extern "C" void kernel_launch(void* const* d_in, const int* in_sizes, int n_in,
                              void* d_out, int out_size, void* d_ws, size_t ws_size,
                              hipStream_t stream) {
    const float* x    = (const float*)d_in[0];   // [1, T, FS, CH]
    const float* Wk   = (const float*)d_in[1];   // [CH, 4U]
    const float* Wr   = (const float*)d_in[2];   // [U, 4U]
    const float* bias = (const float*)d_in[3];   // [4U]
    float* out = (float*)d_out;                  // [1, T, FS, U]

    const size_t bins_bytes = (size_t)TT * FS * sizeof(int);            // ~5.9 MB
    const size_t xz_bytes   = (size_t)TT * FS * NGATES * sizeof(float); // ~1.5 GB
    int*   bins = (int*)d_ws;
    float* xz   = (float*)((char*)d_ws + bins_bytes);
    const bool use_xz = ws_size >= bins_bytes + xz_bytes;   // deterministic per run

    vp_bins_kernel<<<(TT * FS + 255) / 256, 256, 0, stream>>>(x, bins);

    if (use_xz) {
        // Parallel input GEMM across the whole GPU (~1.5GB write, ~66us floor),
        // then a recurrent-only serial chain (half the wmma per step).
        size_t smem_xz = (size_t)(CH * NGATES + NGATES) * sizeof(float);
        vp_xz_kernel<<<TT, 256, smem_xz, stream>>>(x, Wk, bias, xz);
        size_t smem = (size_t)(2 * FSP * UNITS + UNITS * NGATES + NGATES) * sizeof(float)
                    + (size_t)(2 * FSP) * sizeof(int);
        vp_lstm_kernel<true><<<1, 1024, smem, stream>>>(xz, Wk, Wr, bias, bins, out);
    } else {
        size_t smem = (size_t)(2 * FSP * UNITS + 2 * UNITS * NGATES + NGATES) * sizeof(float)
                    + (size_t)(2 * FSP) * sizeof(int);
        vp_lstm_kernel<false><<<1, 1024, smem, stream>>>(x, Wk, Wr, bias, bins, out);
    }
}